// PointerAttention_9414568313115
// MI455X (gfx1250) — compile-verified
//
#include <hip/hip_runtime.h>
#include <cstddef>

typedef __attribute__((ext_vector_type(2))) float v2f;
typedef __attribute__((ext_vector_type(8))) float v8f;

#define B_   4
#define DEC_ 128
#define ENC_ 512
#define H_   768
#define DPB  4           // decoder rows per score block
#define LDA  772         // 768 + 4 pad -> conflict-free LDS (4*m distinct mod 64)

// Hardware V_TANH_F32 (gfx1250 trans op). AMDGCN builtins exist only in the
// device compilation pass, so gate on __HIP_DEVICE_COMPILE__; if the builtin
// is not declared, fall back to raw inline asm for the same instruction
// (device assembler will reject it if the op truly doesn't exist -> signal).
__device__ __forceinline__ float fast_tanh(float x) {
#if defined(__HIP_DEVICE_COMPILE__)
#  if __has_builtin(__builtin_amdgcn_tanhf)
  return __builtin_amdgcn_tanhf(x);
#  else
  float r;
  asm("v_tanh_f32 %0, %1" : "=v"(r) : "v"(x));
  return r;
#  endif
#else
  return tanhf(x);  // host pass: parsed only, never executed
#endif
}

// C(M x 768) = A(M x 768) * W(768 x 768), all fp32 row-major.
// Block = 256 threads = 8 waves. Each block owns one 16-row stripe of A
// (staged in LDS, shared by all 8 waves) and 8 adjacent 16-col tiles of W.
// Each wave computes one 16x16 D tile via V_WMMA_F32_16X16X4_F32, K-loop of 192.
__global__ void wmma_gemm_f32(const float* __restrict__ A,
                              const float* __restrict__ W,
                              float* __restrict__ C, int M) {
  __shared__ float As[16 * LDA];             // 49408 B < 320 KB/WGP
  const int tid  = threadIdx.x;
  const int wid  = tid >> 5;
  const int lane = tid & 31;

  const int nBlocksPerRow = (H_ / 16) / 8;   // 48 n-tiles / 8 waves = 6
  const int tm = blockIdx.x / nBlocksPerRow;
  const int tn = (blockIdx.x % nBlocksPerRow) * 8 + wid;

  // Cooperative, coalesced float4 stage of the 16x768 A stripe into padded LDS.
  const float* Arow = A + (size_t)tm * 16 * H_;
  for (int i = tid; i < 16 * (H_ / 4); i += blockDim.x) {
    int r  = i / (H_ / 4);
    int c4 = i % (H_ / 4);
    float4 v = *(const float4*)(Arow + (size_t)r * H_ + c4 * 4);
    *(float4*)(&As[r * LDA + c4 * 4]) = v;   // (r*772 + 4c)*4 B is 16B-aligned
  }
  __syncthreads();

  const int half  = lane >> 4;               // 0: K=k0..k0+1, 1: K=k0+2..k0+3
  const int ml    = lane & 15;               // M (for A) / N (for B) within tile
  const int khalf = half * 2;

  v8f acc = {};
  const float* Wp = W + (size_t)tn * 16 + ml; // column base, lane-coalesced
#pragma unroll 4
  for (int k0 = 0; k0 < H_; k0 += 4) {
    v2f a = *(const v2f*)(&As[ml * LDA + k0 + khalf]);   // A[M][k0+khalf..+1]
    v2f b;
    b.x = Wp[(size_t)(k0 + khalf) * H_];                 // B[k0+khalf][N]
    b.y = Wp[(size_t)(k0 + khalf + 1) * H_];             // B[k0+khalf+1][N]
    acc = __builtin_amdgcn_wmma_f32_16x16x4_f32(
        /*neg_a=*/false, a, /*neg_b=*/false, b,
        /*c_mod=*/(short)0, acc, /*reuse_a=*/false, /*reuse_b=*/false);
  }

  // D layout: lanes 0-15 -> M = vgpr, lanes 16-31 -> M = vgpr + 8 (ISA 7.12.2)
  const int row0 = tm * 16 + half * 8;
  float* Cp = C + (size_t)row0 * H_ + (size_t)tn * 16 + ml;
#pragma unroll
  for (int i = 0; i < 8; ++i) Cp[(size_t)i * H_] = acc[i];
}

// score[b,d,e] = sum_k vt[k] * tanh(enc_t[b,e,k] + dec_t[b,d,k])
// Block = 256 threads = 8 waves, DPB=4 decoder rows staged in LDS (enc_t reuse x4).
// Each wave owns a strided set of encoder positions; lanes stride K by 32,
// then a 5-step butterfly reduction produces the 4 scores.
__global__ void bahdanau_score_kernel(const float* __restrict__ enc_t,
                                      const float* __restrict__ dec_t,
                                      const float* __restrict__ vt,
                                      const float* __restrict__ mask,
                                      float* __restrict__ out) {
  __shared__ float decs[DPB * H_];
  __shared__ float vts[H_];
  const int tid = threadIdx.x;
  const int gd0 = blockIdx.x * DPB;          // flat (b*DEC + d) of first row
  const int b   = gd0 / DEC_;

  for (int i = tid; i < DPB * H_; i += blockDim.x)
    decs[i] = dec_t[(size_t)gd0 * H_ + i];
  for (int i = tid; i < H_; i += blockDim.x) vts[i] = vt[i];
  __syncthreads();

  const int wid  = tid >> 5;
  const int lane = tid & 31;
  const int NW   = blockDim.x >> 5;          // 8 waves

  for (int e = wid; e < ENC_; e += NW) {
    const float* er = enc_t + ((size_t)b * ENC_ + e) * H_;
    float acc0 = 0.f, acc1 = 0.f, acc2 = 0.f, acc3 = 0.f;
    for (int k = lane; k < H_; k += 32) {    // coalesced 128B enc reads
      float ev = er[k];
      float vv = vts[k];
      acc0 = fmaf(vv, fast_tanh(ev + decs[0 * H_ + k]), acc0);
      acc1 = fmaf(vv, fast_tanh(ev + decs[1 * H_ + k]), acc1);
      acc2 = fmaf(vv, fast_tanh(ev + decs[2 * H_ + k]), acc2);
      acc3 = fmaf(vv, fast_tanh(ev + decs[3 * H_ + k]), acc3);
    }
#pragma unroll
    for (int o = 16; o > 0; o >>= 1) {       // wave32 butterfly reduce
      acc0 += __shfl_xor(acc0, o, 32);
      acc1 += __shfl_xor(acc1, o, 32);
      acc2 += __shfl_xor(acc2, o, 32);
      acc3 += __shfl_xor(acc3, o, 32);
    }
    if (lane == 0) {
      const float accs[DPB] = {acc0, acc1, acc2, acc3};
      const size_t BDE = (size_t)B_ * DEC_ * ENC_;
      for (int dd = 0; dd < DPB; ++dd) {
        size_t idx = (size_t)(gd0 + dd) * ENC_ + e;
        float s = accs[dd];
        out[idx]       = s + mask[idx];      // log_score_masked
        out[BDE + idx] = s;                  // log_score
      }
    }
  }
}

extern "C" void kernel_launch(void* const* d_in, const int* in_sizes, int n_in,
                              void* d_out, int out_size, void* d_ws, size_t ws_size,
                              hipStream_t stream) {
  const float* dec  = (const float*)d_in[0];  // (B, DEC, H)
  const float* enc  = (const float*)d_in[1];  // (B, ENC, H)
  const float* mask = (const float*)d_in[2];  // (B, DEC, ENC)
  const float* W1   = (const float*)d_in[3];  // (H, H)
  const float* W2   = (const float*)d_in[4];  // (H, H)
  const float* vt   = (const float*)d_in[5];  // (H)
  float* out = (float*)d_out;

  float* enc_t = (float*)d_ws;                          // (B*ENC, H) = 6 MB
  float* dec_t = enc_t + (size_t)B_ * ENC_ * H_;        // (B*DEC, H) = 1.5 MB

  const int nBlocksPerRow = (H_ / 16) / 8;              // 6
  // enc_t = enc @ W1 : M = 2048 -> 128 row-stripes * 6 = 768 blocks
  wmma_gemm_f32<<<(B_ * ENC_ / 16) * nBlocksPerRow, 256, 0, stream>>>(
      enc, W1, enc_t, B_ * ENC_);
  // dec_t = dec @ W2 : M = 512 -> 32 * 6 = 192 blocks
  wmma_gemm_f32<<<(B_ * DEC_ / 16) * nBlocksPerRow, 256, 0, stream>>>(
      dec, W2, dec_t, B_ * DEC_);
  // score: 512 decoder rows / 4 per block = 128 blocks
  bahdanau_score_kernel<<<(B_ * DEC_) / DPB, 256, 0, stream>>>(
      enc_t, dec_t, vt, mask, out);
}